// PhotonicDelayReservoir_39350490366784
// MI455X (gfx1250) — compile-verified
//
#include <hip/hip_runtime.h>
#include <cmath>

typedef float v2f __attribute__((ext_vector_type(2)));
typedef float v4f __attribute__((ext_vector_type(4)));
typedef float v8f __attribute__((ext_vector_type(8)));

namespace {
constexpr int kSeq        = 8192;
constexpr int kR          = 80;                 // reservoir dim = 5 tiles of 16
constexpr int kTaps       = 5;
constexpr int kMaxD       = 168;
constexpr int kBPW        = 4;                  // batches per workgroup
constexpr int kNWG        = 64 / kBPW;          // 16 workgroups
constexpr int kBlock      = kTaps * 32;         // 5 waves, one wave per TAP (K-split)
constexpr int kRowStride  = 84;                 // padded floats/row: conflict-free, 16B aligned
constexpr int kSlotStride = kBPW * kRowStride;  // 336 floats / ring slot
constexpr int kRingFloats = kMaxD * kSlotStride;        // 56448 fl = 225,792 B
constexpr int kStgRow     = 84;
constexpr int kStgFloats  = kTaps * 16 * kStgRow;       // tile-4 B stage: 6720 fl
constexpr int kScrSlot    = 32 * 4;                     // 32 lanes * v4f
constexpr int kScrFloats  = kTaps * kTaps * kScrSlot;   // 25 slots = 3200 fl (x2 regions)
constexpr int kLdsFloats  = kRingFloats + kStgFloats + 2 * kScrFloats;  // 69568 fl = 278,272 B
constexpr float kLeak     = 0.3f;
}

__global__ __launch_bounds__(kBlock) __attribute__((amdgpu_waves_per_eu(2)))
void reservoir_wmma_f32(const float* __restrict__ xg,
                        const float* __restrict__ w_in,
                        const float* __restrict__ w_fb,
                        const float* __restrict__ bias,
                        float* __restrict__ out)
{
    extern __shared__ float smem[];
    float* ring = smem;                          // [kMaxD][kBPW][kRowStride]
    float* stg  = smem + kRingFloats;            // [tap][16 rows of tile4][kStgRow]
    float* scrA = stg + kStgFloats;              // partial-C exchange (low half)
    float* scrB = scrA + kScrFloats;             // partial-C exchange (high half)

    const int tid   = threadIdx.x;
    const int wave  = tid >> 5;                  // 0..4 -> this wave's TAP, and epilogue tile
    const int lane  = tid & 31;
    const int l16   = lane & 15;
    const int half4 = (lane < 16) ? 0 : 4;       // WMMA K-half select
    const int iN    = wave * 16 + l16;           // epilogue column (tile = wave)
    const int b0    = blockIdx.x * kBPW;

    // ---- init: zero ring; stage tile-4 B slices; x prefill ----
    for (int idx = tid; idx < kRingFloats; idx += kBlock) ring[idx] = 0.0f;
    for (int idx = tid; idx < kTaps * 16 * kR; idx += kBlock) {
        const int w = idx / (16 * kR);
        const int rem = idx - w * 16 * kR;
        const int r = rem / kR, j = rem - r * kR;
        stg[w * (16 * kStgRow) + r * kStgRow + j] = w_fb[(w * kR + 64 + r) * kR + j];
    }

    // ---- register B fragments: tap = wave, tiles 0..3 (tile 4 comes from LDS stage) ----
    // Fragment (2g):  j = 8g + half4 + {0,1};  (2g+1): j = 8g + half4 + {2,3}
    v2f Bf[4][20];
#pragma unroll
    for (int n = 0; n < 4; ++n) {
#pragma unroll
        for (int f = 0; f < 20; ++f) {
            const int j0 = 8 * (f >> 1) + half4 + 2 * (f & 1);
            Bf[n][f] = *(const v2f*)(w_fb + (wave * kR + 16 * n + l16) * kR + j0);
        }
    }

    const float win = w_in[iN];
    const float bi  = bias[iN];

    float hprev[kBPW] = {0.f, 0.f, 0.f, 0.f};
    float xv[kBPW], xn[kBPW];
#pragma unroll
    for (int b = 0; b < kBPW; ++b) xv[b] = xg[(size_t)(b0 + b) * kSeq];

    const int mrow = l16 & (kBPW - 1);
    const int aoff = mrow * kRowStride + half4;                      // A lane offset in slot
    const int boff = wave * (16 * kStgRow) + l16 * kStgRow + half4;  // staged-B lane offset

    // rolling slots (no per-step modulo)
    constexpr int taps[kTaps] = {1, 4, 24, 96, 168};
    int tapv = taps[0];
    if (wave == 1) tapv = taps[1];
    else if (wave == 2) tapv = taps[2];
    else if (wave == 3) tapv = taps[3];
    else if (wave == 4) tapv = taps[4];
    int slotR = kMaxD - tapv;            // (0 + kMaxD - tap) % kMaxD, tap>=1 so < kMaxD
    if (slotR == kMaxD) slotR = 0;
    int slotW = 0;

    __syncthreads();

    for (int t = 0; t < kSeq; ++t) {
        // software-pipelined next-step input (uniform -> scalar loads)
        const int tn = (t + 1 < kSeq) ? (t + 1) : t;
#pragma unroll
        for (int b = 0; b < kBPW; ++b) xn[b] = xg[(size_t)(b0 + b) * kSeq + tn];

        const float* ap = ring + slotR * kSlotStride + aoff;  // h(t - tap_wave)
        const float* bp = stg + boff;

        // ---- this wave's tap contribution to ALL 5 tiles: 1 A-quad feeds 10 WMMAs ----
        v8f P[kTaps] = {};
#pragma unroll
        for (int g = 0; g < 10; ++g) {
            const v4f aq = *(const v4f*)(ap + 8 * g);   // ds_load_b128 (delayed state)
            const v4f bq = *(const v4f*)(bp + 8 * g);   // ds_load_b128 (tile-4 B)
            const v2f a0 = {aq.x, aq.y}, a1 = {aq.z, aq.w};
            const v2f c0 = {bq.x, bq.y}, c1 = {bq.z, bq.w};
            P[0] = __builtin_amdgcn_wmma_f32_16x16x4_f32(false, a0, false, Bf[0][2*g],   (short)0, P[0], false, false);
            P[1] = __builtin_amdgcn_wmma_f32_16x16x4_f32(false, a0, false, Bf[1][2*g],   (short)0, P[1], false, false);
            P[2] = __builtin_amdgcn_wmma_f32_16x16x4_f32(false, a0, false, Bf[2][2*g],   (short)0, P[2], false, false);
            P[3] = __builtin_amdgcn_wmma_f32_16x16x4_f32(false, a0, false, Bf[3][2*g],   (short)0, P[3], false, false);
            P[4] = __builtin_amdgcn_wmma_f32_16x16x4_f32(false, a0, false, c0,           (short)0, P[4], false, false);
            P[0] = __builtin_amdgcn_wmma_f32_16x16x4_f32(false, a1, false, Bf[0][2*g+1], (short)0, P[0], false, false);
            P[1] = __builtin_amdgcn_wmma_f32_16x16x4_f32(false, a1, false, Bf[1][2*g+1], (short)0, P[1], false, false);
            P[2] = __builtin_amdgcn_wmma_f32_16x16x4_f32(false, a1, false, Bf[2][2*g+1], (short)0, P[2], false, false);
            P[3] = __builtin_amdgcn_wmma_f32_16x16x4_f32(false, a1, false, Bf[3][2*g+1], (short)0, P[3], false, false);
            P[4] = __builtin_amdgcn_wmma_f32_16x16x4_f32(false, a1, false, c1,           (short)0, P[4], false, false);
        }

        // ---- exchange partials: wave w posts its tap's contribution to tiles n != w ----
#pragma unroll
        for (int n = 0; n < kTaps; ++n) {
            if (n != wave) {   // wave-uniform scalar branch
                const v4f lo = __builtin_shufflevector(P[n], P[n], 0, 1, 2, 3);
                const v4f hi = __builtin_shufflevector(P[n], P[n], 4, 5, 6, 7);
                *(v4f*)(scrA + (wave * kTaps + n) * kScrSlot + lane * 4) = lo;
                *(v4f*)(scrB + (wave * kTaps + n) * kScrSlot + lane * 4) = hi;
            }
        }

        // barrier #1: all A reads done (safe to overwrite slot t) + partials visible
        __syncthreads();

        // ---- reduce tile 'wave' over all taps ----
        v8f csum = P[0];
        if (wave == 1) csum = P[1];
        else if (wave == 2) csum = P[2];
        else if (wave == 3) csum = P[3];
        else if (wave == 4) csum = P[4];
#pragma unroll
        for (int w2 = 0; w2 < kTaps; ++w2) {
            if (w2 != wave) {
                const v4f lo = *(const v4f*)(scrA + (w2 * kTaps + wave) * kScrSlot + lane * 4);
                const v4f hi = *(const v4f*)(scrB + (w2 * kTaps + wave) * kScrSlot + lane * 4);
                csum[0] += lo.x; csum[1] += lo.y; csum[2] += lo.z; csum[3] += lo.w;
                csum[4] += hi.x; csum[5] += hi.y; csum[6] += hi.z; csum[7] += hi.w;
            }
        }

        // ---- leaky tanh update + ring/global stores (wave w owns tile w) ----
        if (lane < 16) {           // C rows M=0..7 in lanes 0-15; M=b valid for b<4
#pragma unroll
            for (int b = 0; b < kBPW; ++b) {
                const float hn = (1.0f - kLeak) * hprev[b]
                               + kLeak * tanhf(xv[b] * win + csum[b] + bi);
                hprev[b] = hn;
                ring[slotW * kSlotStride + b * kRowStride + iN] = hn;
                __builtin_nontemporal_store(hn, out + ((size_t)(b0 + b) * kSeq + t) * kR + iN);
            }
        }
#pragma unroll
        for (int b = 0; b < kBPW; ++b) xv[b] = xn[b];

        // barrier #2: slot t fully written before next step's A loads
        __syncthreads();

        slotR = (slotR + 1 == kMaxD) ? 0 : slotR + 1;
        slotW = (slotW + 1 == kMaxD) ? 0 : slotW + 1;
    }
}

extern "C" void kernel_launch(void* const* d_in, const int* in_sizes, int n_in,
                              void* d_out, int out_size, void* d_ws, size_t ws_size,
                              hipStream_t stream) {
    (void)in_sizes; (void)n_in; (void)d_ws; (void)ws_size; (void)out_size;
    const float* x    = (const float*)d_in[0];
    const float* w_in = (const float*)d_in[1];
    const float* w_fb = (const float*)d_in[2];
    const float* bias = (const float*)d_in[3];
    float* out = (float*)d_out;

    const size_t smem = (size_t)kLdsFloats * sizeof(float);  // ~278 KB, fits 320 KB WGP LDS
    static bool attr_set = false;  // idempotent hint, not a stream op (capture-safe)
    if (!attr_set) {
        (void)hipFuncSetAttribute((const void*)reservoir_wmma_f32,
                                  hipFuncAttributeMaxDynamicSharedMemorySize, (int)smem);
        attr_set = true;
    }
    reservoir_wmma_f32<<<kNWG, kBlock, smem, stream>>>(x, w_in, w_fb, bias, out);
}